// GATGraphClassifier4HiddenLayers_38611755991175
// MI455X (gfx1250) — compile-verified
//
#include <hip/hip_runtime.h>

// Fixed by the reference network (hd=32, concat=4*hd, reshape(32,N,4), pool->32)
#define HD      32
#define CAT     128
#define POOL_W  4
#define FEATW   1024   // h_features width = hd*hd

typedef float v2f __attribute__((ext_vector_type(2)));
typedef float v8f __attribute__((ext_vector_type(8)));

__global__ void zero_kernel(float* __restrict__ p, long n) {
    long i = (long)blockIdx.x * blockDim.x + threadIdx.x;
    if (i < n) p[i] = 0.0f;
}

__global__ void degree_kernel(const int* __restrict__ src, const int* __restrict__ dst,
                              int E, float* __restrict__ outdeg, float* __restrict__ indeg) {
    int e = blockIdx.x * blockDim.x + threadIdx.x;
    if (e < E) {
        atomicAdd(&outdeg[src[e]], 1.0f);   // counts are small ints; exact in f32
        atomicAdd(&indeg[dst[e]], 1.0f);
    }
}

__global__ void rsqrt_kernel(float* __restrict__ ns, float* __restrict__ nd, int N) {
    int i = blockIdx.x * blockDim.x + threadIdx.x;
    if (i < N) {
        ns[i] = rsqrtf(fmaxf(ns[i], 1.0f));
        nd[i] = rsqrtf(fmaxf(nd[i], 1.0f));
    }
}

// C(M x 32) = [relu]( (rowScale .* A)(M x K) @ Bm(K x 32) + bias )
// blockDim = 64 (2 waves). Wave w -> columns [16w, 16w+16). Block -> 16 rows.
// Uses V_WMMA_F32_16X16X4_F32 (full fp32, K stepped by 4).
// A fragment (ISA 16x4 f32): lanes 0-15 hold K=k0,k0+1; lanes 16-31 hold K=k0+2,k0+3.
// B fragment mirrors (column-per-lane). C: 8 VGPRs, lanes 0-15 -> M=r, 16-31 -> M=r+8.
__global__ void gemm_wmma_f32(const float* __restrict__ A, int lda,
                              const float* __restrict__ rowScale,
                              const float* __restrict__ Bm,
                              int K, int M,
                              const float* __restrict__ bias, int relu,
                              float* __restrict__ C, int ldc) {
    const int wave = threadIdx.x >> 5;
    const int lane = threadIdx.x & 31;
    const int l16  = lane & 15;
    const int hi   = lane >> 4;            // 0 or 1 (K sub-pair select)
    const int rowBase = blockIdx.x * 16;
    int rA = rowBase + l16; if (rA >= M) rA = M - 1;   // clamp (masked on store)
    const float scale = rowScale ? rowScale[rA] : 1.0f;
    const int col = wave * 16 + l16;
    const float* arow = A + (size_t)rA * lda;

    v8f c = {};
    for (int k0 = 0; k0 < K; k0 += 4) {
        const int kk = k0 + hi * 2;
        v2f a, b;
        a.x = arow[kk]     * scale;
        a.y = arow[kk + 1] * scale;
        b.x = Bm[(size_t)kk       * HD + col];
        b.y = Bm[(size_t)(kk + 1) * HD + col];
        c = __builtin_amdgcn_wmma_f32_16x16x4_f32(false, a, false, b,
                                                  (short)0, c, false, false);
    }
    const float bv = bias ? bias[col] : 0.0f;
#pragma unroll
    for (int r = 0; r < 8; ++r) {
        const int orow = rowBase + r + hi * 8;
        if (orow < M) {
            float v = c[r] + bv;
            if (relu) v = v > 0.0f ? v : 0.0f;
            C[(size_t)orow * ldc + col] = v;
        }
    }
}

// agg[dst] += hp[src] over all edges; 8 threads/edge, 4 channels each (float4 gather).
__global__ void scatter_kernel(const int* __restrict__ src, const int* __restrict__ dst,
                               int E, const float* __restrict__ hp, float* __restrict__ agg) {
    long gid = (long)blockIdx.x * blockDim.x + threadIdx.x;
    int e = (int)(gid >> 3);
    if (e >= E) return;
    int c = (int)(gid & 7) * 4;
    int s = src[e], d = dst[e];
    const float4 v = *(const float4*)(hp + (size_t)s * HD + c);
    float* a = agg + (size_t)d * HD + c;
    atomicAdd(a + 0, v.x);
    atomicAdd(a + 1, v.y);
    atomicAdd(a + 2, v.z);
    atomicAdd(a + 3, v.w);
}

// h_concat[:, off:off+32] = relu(agg * nd + bias)
__global__ void finalize_kernel(const float* __restrict__ agg, const float* __restrict__ nd,
                                const float* __restrict__ bias, float* __restrict__ hcat,
                                int N, int off) {
    long gid = (long)blockIdx.x * blockDim.x + threadIdx.x;
    long n = gid >> 5;
    int  c = (int)(gid & 31);
    if (n < N) {
        float v = agg[n * HD + c] * nd[n] + bias[c];
        hcat[n * CAT + off + c] = v > 0.0f ? v : 0.0f;
    }
}

// hc[a,n,b] = hcat_flat[a*4N + 4n + b]  (raw reshape of (N,128) -> (32,N,4))
// aux[a,i,j] = max over n in [floor(iN/B), ceil((i+1)N/B)) of hc[a,n,j>>3]
// h_features = aux(32,B,32).reshape(B,1024):  row = a*8 + i/32, col = (i%32)*32 + j
__global__ void pool_kernel(const float* __restrict__ hcat, float* __restrict__ hfeat,
                            int N, int B) {
    int t = blockIdx.x * blockDim.x + threadIdx.x;
    int total = 32 * B * POOL_W;
    if (t >= total) return;
    int b = t & 3;
    int i = (t >> 2) % B;
    int a = t / (POOL_W * B);
    long start = ((long)i * N) / B;
    long end   = (((long)(i + 1) * N) + B - 1) / B;
    const float* base = hcat + (size_t)a * 4 * N + b;
    float m = -3.402823466e+38f;
    for (long n = start; n < end; ++n) m = fmaxf(m, base[4 * n]);
    float* o = hfeat + (size_t)(a * 8 + (i >> 5)) * FEATW + (size_t)(i & 31) * 32 + b * 8;
#pragma unroll
    for (int j = 0; j < 8; ++j) o[j] = m;   // width 4 -> 32 is pure replication (j//8)
}

// classification = hrelu(B x 32) @ Wc(32 x nc) + bc
__global__ void classify_kernel(const float* __restrict__ h, const float* __restrict__ Wc,
                                const float* __restrict__ bc, float* __restrict__ out,
                                int B, int nc) {
    int t = blockIdx.x * blockDim.x + threadIdx.x;
    if (t >= B * nc) return;
    int i = t / nc, c = t % nc;
    float acc = bc[c];
#pragma unroll
    for (int j = 0; j < HD; ++j) acc += h[(size_t)i * HD + j] * Wc[j * nc + c];
    out[t] = acc;
}

extern "C" void kernel_launch(void* const* d_in, const int* in_sizes, int n_in,
                              void* d_out, int out_size, void* d_ws, size_t ws_size,
                              hipStream_t stream) {
    const float* x   = (const float*)d_in[0];
    const int*   src = (const int*)d_in[1];
    const int*   dst = (const int*)d_in[2];
    // d_in[3] = num_graphs (device scalar); B derived on host from out_size below.
    const float* W[4]  = {(const float*)d_in[4], (const float*)d_in[6],
                          (const float*)d_in[8], (const float*)d_in[10]};
    const float* bb[4] = {(const float*)d_in[5], (const float*)d_in[7],
                          (const float*)d_in[9], (const float*)d_in[11]};
    const float* Wamp = (const float*)d_in[12];
    const float* bamp = (const float*)d_in[13];
    const float* Wc   = (const float*)d_in[14];
    const float* bc   = (const float*)d_in[15];

    const int in_dim = in_sizes[4] / HD;                  // 128
    const int N      = in_sizes[0] / in_dim;              // 50000
    const int E      = in_sizes[1];                       // 1600000
    const int nc     = in_sizes[14] / HD;                 // 2
    const int B      = (out_size - N * CAT) / (nc + FEATW); // 256

    // d_out layout: classification (B*nc) | h_concat (N*128) | h_features (B*1024)
    float* cls   = (float*)d_out;
    float* hcat  = cls + (size_t)B * nc;
    float* hfeat = hcat + (size_t)N * CAT;

    // workspace: ns | nd | hp | agg | hrelu  (~13.2 MB)
    float* ns   = (float*)d_ws;
    float* nd   = ns + N;
    float* hp   = nd + N;
    float* agg  = hp + (size_t)N * HD;
    float* hrel = agg + (size_t)N * HD;

    auto cdiv = [](long a, long b) { return (int)((a + b - 1) / b); };

    // degrees -> normalization factors (once; graph is fixed across layers)
    zero_kernel<<<cdiv(2L * N, 256), 256, 0, stream>>>(ns, 2L * N);
    degree_kernel<<<cdiv(E, 256), 256, 0, stream>>>(src, dst, E, ns, nd);
    rsqrt_kernel<<<cdiv(N, 256), 256, 0, stream>>>(ns, nd, N);

    for (int L = 0; L < 4; ++L) {
        const float* Ain = (L == 0) ? x : (hcat + (L - 1) * HD);
        const int lda = (L == 0) ? in_dim : CAT;
        const int K   = (L == 0) ? in_dim : HD;
        // hp = (h .* ns) @ W[L]   (fp32 WMMA)
        gemm_wmma_f32<<<cdiv(N, 16), 64, 0, stream>>>(Ain, lda, ns, W[L], K, N,
                                                      nullptr, 0, hp, HD);
        zero_kernel<<<cdiv((long)N * HD, 256), 256, 0, stream>>>(agg, (long)N * HD);
        scatter_kernel<<<cdiv((long)E * 8, 256), 256, 0, stream>>>(src, dst, E, hp, agg);
        finalize_kernel<<<cdiv((long)N * HD, 256), 256, 0, stream>>>(agg, nd, bb[L],
                                                                     hcat, N, L * HD);
    }

    pool_kernel<<<cdiv(32L * B * POOL_W, 256), 256, 0, stream>>>(hcat, hfeat, N, B);
    // hrelu = relu(h_features @ Wamp + bamp)   (fp32 WMMA, K=1024)
    gemm_wmma_f32<<<cdiv(B, 16), 64, 0, stream>>>(hfeat, FEATW, nullptr, Wamp, FEATW, B,
                                                  bamp, 1, hrel, HD);
    classify_kernel<<<cdiv((long)B * nc, 256), 256, 0, stream>>>(hrel, Wc, bc, cls, B, nc);
}